// SpikingAttention_69724499083935
// MI455X (gfx1250) — compile-verified
//
#include <hip/hip_runtime.h>
#include <hip/hip_bf16.h>
#include <math.h>

// ---------------------------------------------------------------------------
// SpikingAttention on MI455X (gfx1250, wave32)
//   B=2, S=8192, D=128, WINDOW=128, STRIDE=64, NUM_GLOBAL=4, TAU_MEM=10
// ---------------------------------------------------------------------------

#define S_LEN   8192
#define D_DIM   128
#define B_BATCH 2
#define WINDOW  128
#define STRIDE_G 64

typedef float v2f __attribute__((ext_vector_type(2)));
typedef float v8f __attribute__((ext_vector_type(8)));
typedef int   sa_v4i __attribute__((vector_size(16)));   // 16B granule for async LDS copy

#if defined(__HIP_DEVICE_COMPILE__) && __has_builtin(__builtin_amdgcn_global_load_async_to_lds_b128)
#define SA_HAVE_ASYNC_LDS 1
#else
#define SA_HAVE_ASYNC_LDS 0
#endif

__device__ __forceinline__ void sa_wait_async0() {
#if defined(__HIP_DEVICE_COMPILE__)
#if __has_builtin(__builtin_amdgcn_s_wait_asynccnt)
    __builtin_amdgcn_s_wait_asynccnt(0);
#else
    asm volatile("s_wait_asynccnt 0x0" ::: "memory");
#endif
#endif
}

// ---------------------------------------------------------------------------
// Kernel 1: input current  cur[b,s] = x[b,s,:] . Ws + bs
// ---------------------------------------------------------------------------
__global__ void sa_cur_kernel(const float* __restrict__ x,
                              const float* __restrict__ Ws,
                              const float* __restrict__ bs,
                              float* __restrict__ cur, int n) {
    int idx = blockIdx.x * blockDim.x + threadIdx.x;
    if (idx >= n) return;
    const float* xp = x + (long)idx * D_DIM;
    float s = 0.f;
#pragma unroll 8
    for (int d = 0; d < D_DIM; ++d) s = fmaf(xp[d], Ws[d], s);
    cur[idx] = s + bs[0];
}

// ---------------------------------------------------------------------------
// Kernel 2: sequential LIF scan (hard reset), then OR over batch -> spike mask
// ---------------------------------------------------------------------------
__global__ void sa_lif_kernel(const float* __restrict__ cur,
                              float* __restrict__ spkbuf,
                              int* __restrict__ spike) {
    const float decay = 0.90483741803595952f;   // exp(-1/10)
    int t = threadIdx.x;
    if (t < B_BATCH) {
        float mem = 0.f;
        const float* c = cur + (long)t * S_LEN;
        float* sp = spkbuf + (long)t * S_LEN;
        for (int s = 0; s < S_LEN; ++s) {
            mem = fmaf(decay, mem, c[s]);
            float spk = (mem >= 1.0f) ? 1.f : 0.f;
            if (spk != 0.f) mem = 0.f;
            sp[s] = spk;
        }
    }
    __syncthreads();
    for (int s = t; s < S_LEN; s += blockDim.x)
        spike[s] = (spkbuf[s] != 0.f || spkbuf[S_LEN + s] != 0.f) ? 1 : 0;
}

// ---------------------------------------------------------------------------
// Kernel 3: GEMM  C[M x 128] = A[M x 128] * W[128 x 128] + bias
// Block = 256 threads = 8 waves; one workgroup owns a 16-row stripe.
// Step 1: async-copy W (64 KB) global -> LDS (ASYNCcnt path), wait + barrier.
// Step 2: wave w computes the 16x16 tile at columns [16w, 16w+16) with
//         V_WMMA_F32_16X16X4_F32; B-fragments stream from LDS (ds_load),
//         A-fragments from global (L2-resident).
// A frag (16x4):  lanes 0-15 hold K = k+0,k+1 ; lanes 16-31 hold K = k+2,k+3
// B frag (4x16):  VGPR0/1 hold rows K = k+2*half / +1, N = lane&15
// C/D (16x16):    VGPR r: lanes 0-15 -> M=r, lanes 16-31 -> M=r+8, N = lane&15
// ---------------------------------------------------------------------------
__global__ void sa_wmma_gemm128(const float* __restrict__ A,
                                const float* __restrict__ W,
                                const float* __restrict__ bias,
                                float* __restrict__ C) {
    __shared__ float Wlds[D_DIM * D_DIM];   // 64 KB of the 320 KB WGP LDS

    const int tid  = threadIdx.x;           // 0..255
    const int wave = tid >> 5;              // 0..7  -> column tile
    const int lane = tid & 31;
    const int half = lane >> 4;             // 0 or 1
    const int idx  = lane & 15;
    const long row0 = (long)blockIdx.x * 16;
    const int  col0 = wave * 16;

    // ---- stage W into LDS: 16384 floats = 4096 x 16B granules; 16/thread ----
#if SA_HAVE_ASYNC_LDS
#pragma unroll
    for (int t = 0; t < 16; ++t) {
        int e = (tid + t * 256) * 4;        // float index, 16B granules
        __builtin_amdgcn_global_load_async_to_lds_b128(
            (__attribute__((address_space(1))) sa_v4i*)(W + e),
            (__attribute__((address_space(3))) sa_v4i*)(Wlds + e),
            0, 0);
    }
    sa_wait_async0();
#else
#pragma unroll
    for (int t = 0; t < 16; ++t) {
        int e = tid + t * 256;
        ((float4*)Wlds)[e] = ((const float4*)W)[e];
    }
#endif
    __syncthreads();

    // ---- 16x16 tile accumulate over K=128 in steps of 4 ----
    v8f c = {};
    const float* arow = A + (row0 + idx) * (long)D_DIM + half * 2;
    const float* brow = Wlds + (half * 2) * D_DIM + col0 + idx;
#pragma unroll 4
    for (int k = 0; k < D_DIM; k += 4) {
        v2f a, b;
        a[0] = arow[k];
        a[1] = arow[k + 1];
        const float* bp = brow + k * D_DIM;
        b[0] = bp[0];
        b[1] = bp[D_DIM];
        c = __builtin_amdgcn_wmma_f32_16x16x4_f32(
                false, a, false, b, (short)0, c, false, false);
    }

    float bv = bias[col0 + idx];
#pragma unroll
    for (int r = 0; r < 8; ++r) {
        int m = r + half * 8;
        C[(row0 + m) * (long)D_DIM + col0 + idx] = c[r] + bv;
    }
}

// ---------------------------------------------------------------------------
// Kernel 4: sparse masked online-softmax attention, one wave per query row.
// Candidate columns (deduplicated):
//   local:  j in [max(0, i-128), i]
//   global: j % 64 == 0, outside the local range
//   first:  j in {1,2,3} (j=0 covered by globals), outside local range
// all gated by spike[j].  Fully-masked rows output 0 (nan_to_num semantics).
// ---------------------------------------------------------------------------
__global__ void sa_attn_kernel(const float* __restrict__ Q,
                               const float* __restrict__ K,
                               const float* __restrict__ V,
                               const int* __restrict__ spike,
                               float* __restrict__ O) {
    const float scale = 0.088388347648318447f;   // 1/sqrt(128)
    const int lane = threadIdx.x & 31;
    const int wave = threadIdx.x >> 5;
    const long row = (long)blockIdx.x * (blockDim.x >> 5) + wave;
    const int b = (int)(row >> 13);        // row / 8192
    const int i = (int)(row & (S_LEN - 1));

    const float4 q = ((const float4*)(Q + row * D_DIM))[lane];
    const float* Kb = K + (long)b * S_LEN * D_DIM;
    const float* Vb = V + (long)b * S_LEN * D_DIM;

    float m = -INFINITY, l = 0.f;
    float4 acc = make_float4(0.f, 0.f, 0.f, 0.f);

    auto process = [&](int j) {
        const float* vrow = Vb + (long)j * D_DIM;
        __builtin_prefetch(vrow + lane * 4, 0, 0);   // global_prefetch_b8
        float4 kv = ((const float4*)(Kb + (long)j * D_DIM))[lane];
        float p = q.x * kv.x + q.y * kv.y + q.z * kv.z + q.w * kv.w;
#pragma unroll
        for (int off = 16; off > 0; off >>= 1)
            p += __shfl_xor(p, off, 32);             // wave32 reduction
        float s = p * scale;
        float mn = fmaxf(m, s);
        float corr = __expf(m - mn);                 // 0 on first candidate
        float w = __expf(s - mn);
        float4 vv = ((const float4*)vrow)[lane];
        acc.x = fmaf(acc.x, corr, w * vv.x);
        acc.y = fmaf(acc.y, corr, w * vv.y);
        acc.z = fmaf(acc.z, corr, w * vv.z);
        acc.w = fmaf(acc.w, corr, w * vv.w);
        l = fmaf(l, corr, w);
        m = mn;
    };

    int lo = i - WINDOW; if (lo < 0) lo = 0;
    for (int j = lo; j <= i; ++j)
        if (spike[j]) process(j);
    for (int j = 0; j < S_LEN; j += STRIDE_G)
        if ((j < lo || j > i) && spike[j]) process(j);
    for (int j = 1; j < 4; ++j)
        if ((j < lo || j > i) && spike[j]) process(j);

    float inv = (l > 0.f) ? 1.f / l : 0.f;
    float4 o = make_float4(acc.x * inv, acc.y * inv, acc.z * inv, acc.w * inv);
    ((float4*)(O + row * D_DIM))[lane] = o;
}

// ---------------------------------------------------------------------------
// Host-side launcher
// Inputs (setup_inputs order): x, Wq, bq, Wk, bk, Wv, bv, Wo, bo, Ws, bs
// ---------------------------------------------------------------------------
extern "C" void kernel_launch(void* const* d_in, const int* in_sizes, int n_in,
                              void* d_out, int out_size, void* d_ws, size_t ws_size,
                              hipStream_t stream) {
    const float* x  = (const float*)d_in[0];
    const float* Wq = (const float*)d_in[1];
    const float* bq = (const float*)d_in[2];
    const float* Wk = (const float*)d_in[3];
    const float* bk = (const float*)d_in[4];
    const float* Wv = (const float*)d_in[5];
    const float* bv = (const float*)d_in[6];
    const float* Wo = (const float*)d_in[7];
    const float* bo = (const float*)d_in[8];
    const float* Ws = (const float*)d_in[9];
    const float* bs = (const float*)d_in[10];
    float* out = (float*)d_out;

    const long N = (long)B_BATCH * S_LEN;   // 16384 rows

    // Workspace carve-out (all 16B-aligned; total ~32.2 MB)
    char* p = (char*)d_ws;
    float* cur    = (float*)p; p += N * sizeof(float);         // 64 KB
    float* spkbuf = (float*)p; p += N * sizeof(float);         // 64 KB
    int*   spike  = (int*)p;   p += S_LEN * sizeof(int);       // 32 KB
    float* Qm     = (float*)p; p += N * D_DIM * sizeof(float); // 8 MB
    float* Km     = (float*)p; p += N * D_DIM * sizeof(float); // 8 MB
    float* Vm     = (float*)p; p += N * D_DIM * sizeof(float); // 8 MB
    float* Om     = (float*)p; p += N * D_DIM * sizeof(float); // 8 MB

    // 1) input current
    sa_cur_kernel<<<dim3((N + 255) / 256), dim3(256), 0, stream>>>(x, Ws, bs, cur, (int)N);

    // 2) sequential LIF scan + batch-OR spike mask (single workgroup)
    sa_lif_kernel<<<dim3(1), dim3(256), 0, stream>>>(cur, spkbuf, spike);

    // 3) Q/K/V projections: 1 block per 16-row stripe, 8 waves = 8 col tiles
    dim3 gemm_grid((unsigned)(N / 16));
    sa_wmma_gemm128<<<gemm_grid, dim3(256), 0, stream>>>(x, Wq, bq, Qm);
    sa_wmma_gemm128<<<gemm_grid, dim3(256), 0, stream>>>(x, Wk, bk, Km);
    sa_wmma_gemm128<<<gemm_grid, dim3(256), 0, stream>>>(x, Wv, bv, Vm);

    // 4) sparse masked attention: 4 waves / block, one wave per query row
    sa_attn_kernel<<<dim3((unsigned)(N / 4)), dim3(128), 0, stream>>>(Qm, Km, Vm, spike, Om);

    // 5) output projection
    sa_wmma_gemm128<<<gemm_grid, dim3(256), 0, stream>>>(Om, Wo, bo, out);
}